// CausalSelfAttention_37812892074090
// MI455X (gfx1250) — compile-verified
//
#include <hip/hip_runtime.h>
#include <hip/hip_bf16.h>

typedef float v2f __attribute__((ext_vector_type(2)));
typedef float v8f __attribute__((ext_vector_type(8)));

#define WMMA_F32(a, b, c) \
  __builtin_amdgcn_wmma_f32_16x16x4_f32(false, (a), false, (b), (short)0, (c), false, false)

// ---- CDNA5 async global->LDS copy (ASYNCcnt-tracked), with safe fallback ----
#if defined(__HIP_DEVICE_COMPILE__) && __has_builtin(__builtin_amdgcn_global_load_async_to_lds_b128)
#define ASYNC_CP 1
#else
#define ASYNC_CP 0
#endif

typedef int i32x4 __attribute__((__vector_size__(16)));
typedef __attribute__((address_space(1))) i32x4 as1_i32x4;
typedef __attribute__((address_space(3))) i32x4 as3_i32x4;

__device__ __forceinline__ void cp_async_b128(const float* g, float* l) {
#if ASYNC_CP
  __builtin_amdgcn_global_load_async_to_lds_b128(
      (as1_i32x4*)g, (as3_i32x4*)l, 0, 0);
#else
  *(float4*)l = *(const float4*)g;
#endif
}

#if ASYNC_CP && __has_builtin(__builtin_amdgcn_s_wait_asynccnt)
#define WAIT_ASYNC(n) __builtin_amdgcn_s_wait_asynccnt(n)
#elif ASYNC_CP
#define WAIT_ASYNC(n) asm volatile("s_wait_asynccnt " #n ::: "memory")
#else
#define WAIT_ASYNC(n) do {} while (0)
#endif

// ---------------------------------------------------------------------------
// Tiled f32 GEMM with V_WMMA_F32_16X16X4_F32, double-buffered async staging.
//   C[M,N] = A[M,K] @ W[K,N] + bias[N]
// mode 0: scatter columns as Q/K/V -> out0/out1/out2 laid out [B,H,S,Hd]
// mode 1: plain row-major store to out0[M,N]
// Block tile 128x128, K-chunk 32, 256 threads = 8 waves, wave tile 64x32.
// ---------------------------------------------------------------------------
__global__ __launch_bounds__(256) void gemm_f32_wmma(
    const float* __restrict__ A, const float* __restrict__ W,
    const float* __restrict__ bias,
    float* __restrict__ out0, float* __restrict__ out1, float* __restrict__ out2,
    int M, int N, int K, int mode)
{
  constexpr int BM = 128, BN = 128, KC = 32;
  constexpr int LDA = KC + 4;   // 36 floats = 144 B rows (16B aligned)
  constexpr int LDB = BN + 4;   // 132 floats = 528 B rows (16B aligned)
  constexpr int ASZ = BM * LDA, BSZ = KC * LDB;
  __shared__ __align__(16) float As[2 * ASZ];  // As[m][k]
  __shared__ __align__(16) float Bs[2 * BSZ];  // Bs[k][n] (natural layout)

  const int tid   = threadIdx.x;
  const int lane  = tid & 31;
  const int wid   = tid >> 5;
  const int m0    = blockIdx.y * BM;
  const int n0    = blockIdx.x * BN;
  const int wm    = (wid >> 2) * 64;  // wave m-offset within block tile
  const int wn    = (wid & 3) * 32;   // wave n-offset within block tile
  const int lhalf = lane >> 4;        // selects K pair (ISA 16x4 A layout)
  const int l16   = lane & 15;        // M (A) / N (B) index within fragment

  // stage one K-chunk into buffer `buf`: 8 async B128 copies per thread
  auto stage = [&](int k0, int buf) {
    float* as = As + buf * ASZ;
    float* bs = Bs + buf * BSZ;
    for (int p = 0; p < 4; ++p) {               // A tile: 128 x 32
      const int L = tid + p * 256;
      const int m = L >> 3, kc = (L & 7) * 4;
      cp_async_b128(A + (size_t)(m0 + m) * K + k0 + kc, as + m * LDA + kc);
    }
    for (int p = 0; p < 4; ++p) {               // W tile: 32 x 128 (coalesced in n)
      const int L = tid + p * 256;
      const int k = L >> 5, nc = (L & 31) * 4;
      cp_async_b128(W + (size_t)(k0 + k) * N + n0 + nc, bs + k * LDB + nc);
    }
  };

  v8f acc[4][2] = {};

  stage(0, 0);
  int buf = 0;
  for (int k0 = 0; k0 < K; k0 += KC, buf ^= 1) {
    const bool more = (k0 + KC) < K;
    if (more) stage(k0 + KC, buf ^ 1);
    if (more) { WAIT_ASYNC(8); } else { WAIT_ASYNC(0); }
    __syncthreads();   // current buffer fully resident for all waves

    const float* as = As + buf * ASZ;
    const float* bs = Bs + buf * BSZ;
    for (int ks = 0; ks < KC; ks += 4) {
      const int kk = ks + 2 * lhalf;
      v2f a[4], b[2];
      for (int im = 0; im < 4; ++im)
        a[im] = *(const v2f*)&as[(wm + im * 16 + l16) * LDA + kk];
      for (int in = 0; in < 2; ++in) {
        const int nn = wn + in * 16 + l16;
        b[in].x = bs[kk * LDB + nn];
        b[in].y = bs[(kk + 1) * LDB + nn];
      }
      for (int im = 0; im < 4; ++im)
        for (int in = 0; in < 2; ++in)
          acc[im][in] = WMMA_F32(a[im], b[in], acc[im][in]);
    }
    __syncthreads();   // all waves done with `buf` before it is restaged
  }

  // --- epilogue: C/D layout => row = lhalf*8 + i, col = l16 ---
  for (int im = 0; im < 4; ++im) {
    for (int in = 0; in < 2; ++in) {
      for (int i = 0; i < 8; ++i) {
        const int m = m0 + wm + im * 16 + lhalf * 8 + i;
        const int n = n0 + wn + in * 16 + l16;
        const float v = acc[im][in][i] + bias[n];
        if (mode == 0) {
          // split 2304 cols into Q/K/V, reshape [B*S, H*Hd] -> [B,H,S,Hd]
          const int sel = n / 768;
          const int d   = n - sel * 768;
          const int h   = d >> 6;
          const int hd  = d & 63;
          const int bb  = m >> 11;      // S = 2048
          const int ss  = m & 2047;
          float* dst = (sel == 0) ? out0 : ((sel == 1) ? out1 : out2);
          dst[(((size_t)(bb * 12 + h)) * 2048 + ss) * 64 + hd] = v;
        } else {
          out0[(size_t)m * N + n] = v;
        }
      }
    }
  }
}

// ---------------------------------------------------------------------------
// Flash attention (causal, f32 WMMA), double-buffered async K/V staging.
// Grid: (S/128, H, B). Block: 256. Q/K/V: [B,H,S,Hd]. ctx: [B,S,H*Hd].
// Each wave owns a 16-row q strip.
// ---------------------------------------------------------------------------
__global__ __launch_bounds__(256) void flash_attn_f32_wmma(
    const float* __restrict__ Q, const float* __restrict__ Kv,
    const float* __restrict__ V, float* __restrict__ ctx)
{
  constexpr int S = 2048, H = 12, Hd = 64, BQ = 128, BKV = 64;
  constexpr int LDQ = Hd + 4;    // 68 floats = 272 B rows (16B aligned)
  constexpr int LDK = Hd + 4;
  constexpr int LDP = BKV + 2;
  constexpr int KSZ = BKV * LDK;
  __shared__ __align__(16) float Qs[BQ * LDQ];       // Qs[m][hd]
  __shared__ __align__(16) float Ks[2 * KSZ];        // Ks[kv][hd]
  __shared__ __align__(16) float Vs[2 * KSZ];        // Vs[kv][hd] (natural)
  __shared__ __align__(16) float Ps[8 * 16 * LDP];   // wave-private P strips

  const int tid   = threadIdx.x;
  const int lane  = tid & 31;
  const int wid   = tid >> 5;
  const int lhalf = lane >> 4;
  const int l16   = lane & 15;
  const int q0    = blockIdx.x * BQ;
  const int h     = blockIdx.y;
  const int b     = blockIdx.z;
  const int strip = wid * 16;
  const size_t base = ((size_t)(b * H + h)) * S * Hd;
  const float scale = 0.125f;  // 64^-0.5

  auto stage_kv = [&](int j0, int buf) {   // 8 async B128 copies per thread
    float* ks = Ks + buf * KSZ;
    float* vs = Vs + buf * KSZ;
    for (int p = 0; p < 4; ++p) {
      const int L = tid + p * 256;
      const int n = L >> 4, c = (L & 15) * 4;
      cp_async_b128(Kv + base + (size_t)(j0 + n) * Hd + c, ks + n * LDK + c);
    }
    for (int p = 0; p < 4; ++p) {
      const int L = tid + p * 256;
      const int n = L >> 4, c = (L & 15) * 4;
      cp_async_b128(V + base + (size_t)(j0 + n) * Hd + c, vs + n * LDK + c);
    }
  };

  // prologue: stage Q tile (8 copies/thread) + first K/V tile pair
  for (int p = 0; p < 8; ++p) {
    const int L = tid + p * 256;
    const int m = L >> 4, c = (L & 15) * 4;
    cp_async_b128(Q + base + (size_t)(q0 + m) * Hd + c, Qs + m * LDQ + c);
  }
  stage_kv(0, 0);

  v8f O[4] = {};
  float row_m[8], row_l[8];
  for (int i = 0; i < 8; ++i) { row_m[i] = -1e30f; row_l[i] = 0.0f; }

  const int nj = (q0 + BQ) / BKV;  // causal: tiles at or left of the diagonal
  int buf = 0;
  for (int jt = 0; jt < nj; ++jt, buf ^= 1) {
    const int j0 = jt * BKV;
    const bool more = (jt + 1) < nj;
    if (more) stage_kv(j0 + BKV, buf ^ 1);
    if (more) { WAIT_ASYNC(8); } else { WAIT_ASYNC(0); }
    __syncthreads();

    const float* ks = Ks + buf * KSZ;
    const float* vs = Vs + buf * KSZ;

    // ---- S strip = Q_strip @ K^T : 4 tiles x 16 chained K=4 WMMAs ----
    v8f sacc[4] = {};
    for (int kd = 0; kd < Hd; kd += 4) {
      const int kk = kd + 2 * lhalf;
      const v2f aq = *(const v2f*)&Qs[(strip + l16) * LDQ + kk];
      for (int t = 0; t < 4; ++t) {
        const v2f bk = *(const v2f*)&ks[(t * 16 + l16) * LDK + kk];
        sacc[t] = WMMA_F32(aq, bk, sacc[t]);
      }
    }

    // ---- scale + causal mask + row max (reduce across 16-lane half) ----
    float tmax[8];
    for (int i = 0; i < 8; ++i) tmax[i] = -1e30f;
    for (int t = 0; t < 4; ++t) {
      const int colg = j0 + t * 16 + l16;
      for (int i = 0; i < 8; ++i) {
        const int rowg = q0 + strip + lhalf * 8 + i;
        float s = sacc[t][i] * scale;
        s = (colg <= rowg) ? s : -1e30f;
        sacc[t][i] = s;
        tmax[i] = fmaxf(tmax[i], s);
      }
    }
    for (int i = 0; i < 8; ++i)
      for (int off = 1; off < 16; off <<= 1)
        tmax[i] = fmaxf(tmax[i], __shfl_xor(tmax[i], off, 16));

    // ---- online softmax update ----
    float corr[8], psum[8];
    for (int i = 0; i < 8; ++i) {
      const float mnew = fmaxf(row_m[i], tmax[i]);
      corr[i] = __expf(row_m[i] - mnew);
      row_m[i] = mnew;
      psum[i] = 0.0f;
    }
    for (int t = 0; t < 4; ++t)
      for (int i = 0; i < 8; ++i) {
        const float p = __expf(sacc[t][i] - row_m[i]);
        sacc[t][i] = p;
        psum[i] += p;
      }
    for (int i = 0; i < 8; ++i) {
      for (int off = 1; off < 16; off <<= 1)
        psum[i] += __shfl_xor(psum[i], off, 16);
      row_l[i] = row_l[i] * corr[i] + psum[i];
      for (int t = 0; t < 4; ++t) O[t][i] = O[t][i] * corr[i];
    }

    // ---- P strip -> wave-private LDS (C-layout -> A-layout bounce) ----
    float* P = Ps + wid * 16 * LDP;
    for (int t = 0; t < 4; ++t)
      for (int i = 0; i < 8; ++i)
        P[(lhalf * 8 + i) * LDP + t * 16 + l16] = sacc[t][i];

    // ---- O += P @ V : 4 tiles x 16 chained K=4 WMMAs ----
    for (int kk0 = 0; kk0 < BKV; kk0 += 4) {
      const int kk = kk0 + 2 * lhalf;
      const v2f ap = *(const v2f*)&P[l16 * LDP + kk];
      for (int t = 0; t < 4; ++t) {
        const int col = t * 16 + l16;
        v2f bv;
        bv.x = vs[kk * LDK + col];
        bv.y = vs[(kk + 1) * LDK + col];
        O[t] = WMMA_F32(ap, bv, O[t]);
      }
    }
    __syncthreads();   // all waves done with `buf` before it is restaged
  }

  // ---- normalize + store ctx[B,S,H*Hd] ----
  for (int t = 0; t < 4; ++t)
    for (int i = 0; i < 8; ++i) {
      const int rowg = q0 + strip + lhalf * 8 + i;
      const float v = O[t][i] / row_l[i];
      ctx[((size_t)(b * S + rowg) * H + h) * Hd + t * 16 + l16] = v;
    }
}

// ---------------------------------------------------------------------------
extern "C" void kernel_launch(void* const* d_in, const int* in_sizes, int n_in,
                              void* d_out, int out_size, void* d_ws, size_t ws_size,
                              hipStream_t stream) {
  const float* x     = (const float*)d_in[0];   // [4,2048,768]
  const float* qkv_w = (const float*)d_in[1];   // [768,2304]
  const float* qkv_b = (const float*)d_in[2];   // [2304]
  const float* out_w = (const float*)d_in[3];   // [768,768]
  const float* out_b = (const float*)d_in[4];   // [768]
  float* out = (float*)d_out;                   // [4,2048,768]
  float* ws  = (float*)d_ws;

  const size_t per = (size_t)4 * 12 * 2048 * 64;  // one [B,H,S,Hd] tensor
  float* q_ws = ws;
  float* k_ws = ws + per;
  float* v_ws = ws + 2 * per;
  float* ctx  = ws + 3 * per;

  dim3 blk(256);
  // 1) QKV projection + head split/transpose (M=8192, N=2304, K=768)
  gemm_f32_wmma<<<dim3(2304 / 128, 8192 / 128), blk, 0, stream>>>(
      x, qkv_w, qkv_b, q_ws, k_ws, v_ws, 8192, 2304, 768, 0);
  // 2) causal flash attention
  flash_attn_f32_wmma<<<dim3(2048 / 128, 12, 4), blk, 0, stream>>>(
      q_ws, k_ws, v_ws, ctx);
  // 3) output projection (M=8192, N=768, K=768)
  gemm_f32_wmma<<<dim3(768 / 128, 8192 / 128), blk, 0, stream>>>(
      ctx, out_w, out_b, out, nullptr, nullptr, 8192, 768, 768, 1);
}